// EfficientDeformablePooling_74938589381106
// MI455X (gfx1250) — compile-verified
//
#include <hip/hip_runtime.h>
#include <hip/hip_bf16.h>

typedef __attribute__((ext_vector_type(2))) float v2f;
typedef __attribute__((ext_vector_type(8))) float v8f;

#define NSP 8
#define NC  64
#define NB  2
#define NP  32
#define NH  64
#define NW  64
#define NHW 4096
#define NL  3

// -------- Kernel 1: 1x1 offset conv: off[b][ch][hw], ch in 0..15 --------
__global__ void __launch_bounds__(256)
edp_offset_conv(const float* __restrict__ feat0,
                const float* __restrict__ w_off,
                const float* __restrict__ b_off,
                float* __restrict__ off) {
  int idx = blockIdx.x * blockDim.x + threadIdx.x;   // 0 .. B*HW-1
  int hw  = idx & (NHW - 1);
  int b   = idx >> 12;
  float acc[16];
#pragma unroll
  for (int ch = 0; ch < 16; ++ch) acc[ch] = b_off[ch];
  const float* fb = feat0 + (size_t)b * NC * NHW + hw;
  for (int c = 0; c < NC; ++c) {
    float f = fb[(size_t)c * NHW];
#pragma unroll
    for (int ch = 0; ch < 16; ++ch)
      acc[ch] = fmaf(w_off[ch * NC + c], f, acc[ch]);
  }
#pragma unroll
  for (int ch = 0; ch < 16; ++ch)
    off[(size_t)(b * 16 + ch) * NHW + hw] = acc[ch];
}

// -------- Kernel 2: deformable bilinear sampling (zero pad) --------
// S layout: S[((l*NB + b)*NC + c)*NHW + hw]  (K-major for the GEMM)
__global__ void __launch_bounds__(256)
edp_deform_sample(const float* __restrict__ f0,
                  const float* __restrict__ f1,
                  const float* __restrict__ f2,
                  const float* __restrict__ off,
                  float* __restrict__ S) {
  int idx = blockIdx.x * blockDim.x + threadIdx.x;   // 0 .. L*B*SP*HW-1
  int hw = idx & (NHW - 1);
  int g  = (idx >> 12) & 7;
  int b  = (idx >> 15) & 1;
  int l  = idx >> 16;
  const float* feat = (l == 0) ? f0 : (l == 1) ? f1 : f2;

  float dy = off[(size_t)(b * 16 + 2 * g) * NHW + hw];
  float dx = off[(size_t)(b * 16 + 2 * g + 1) * NHW + hw];
  float y = (float)(hw >> 6) + dy;
  float x = (float)(hw & 63) + dx;
  float y0f = floorf(y), x0f = floorf(x);
  int y0 = (int)y0f, x0 = (int)x0f;
  float wy1 = y - y0f, wx1 = x - x0f;
  float wy0 = 1.0f - wy1, wx0 = 1.0f - wx1;

  int y0c = min(max(y0, 0), NH - 1), y1c = min(max(y0 + 1, 0), NH - 1);
  int x0c = min(max(x0, 0), NW - 1), x1c = min(max(x0 + 1, 0), NW - 1);
  bool vy0 = (y0 >= 0) & (y0 <= NH - 1);
  bool vy1 = (y0 + 1 >= 0) & (y0 + 1 <= NH - 1);
  bool vx0 = (x0 >= 0) & (x0 <= NW - 1);
  bool vx1 = (x0 + 1 >= 0) & (x0 + 1 <= NW - 1);

  float w00 = wy0 * wx0 * (float)(vy0 & vx0);
  float w01 = wy0 * wx1 * (float)(vy0 & vx1);
  float w10 = wy1 * wx0 * (float)(vy1 & vx0);
  float w11 = wy1 * wx1 * (float)(vy1 & vx1);
  int o00 = y0c * NW + x0c, o01 = y0c * NW + x1c;
  int o10 = y1c * NW + x0c, o11 = y1c * NW + x1c;

  size_t sbase = ((size_t)(l * NB + b) * NC) * NHW + hw;
#pragma unroll
  for (int j = 0; j < 8; ++j) {
    int c = g * 8 + j;
    const float* fc = feat + (size_t)(b * NC + c) * NHW;
    float v = fc[o00] * w00 + fc[o01] * w01 + fc[o10] * w10 + fc[o11] * w11;
    S[sbase + (size_t)c * NHW] = v;
  }
}

// -------- Kernel 3: f32 WMMA GEMM (64x64 weight x 64x4096 sampled) --------
// Block = 4 waves; wave w owns M-tile w (16 rows). Block owns one (l,b,ntile).
// Output broadcast over P=32 priors with non-temporal streaming stores.
__global__ void __launch_bounds__(128)
edp_gemm_wmma(const float* __restrict__ S,
              const float* __restrict__ weight,
              const float* __restrict__ bias,
              float* __restrict__ out) {
  const int lane  = threadIdx.x & 31;
  const int mtile = threadIdx.x >> 5;        // 0..3
  const int ntile = blockIdx.x & 255;        // 0..255
  const int lb    = blockIdx.x >> 8;         // 0..5 = l*NB + b
  const int l = lb >> 1, b = lb & 1;

  const int m0    = mtile << 4;
  const int n0    = ntile << 4;
  const int pix   = n0 + (lane & 15);
  const int khalf = (lane >> 4) << 1;        // 0 (lanes 0-15) or 2 (lanes 16-31)
  const int mrow  = m0 + (lane & 15);

  const float* Sb = S + (size_t)lb * NC * NHW + pix;
  const float* Wa = weight + mrow * NC;

  v8f acc = {};
#pragma unroll
  for (int k0 = 0; k0 < NC; k0 += 4) {
    v2f a, bf;
    a.x  = Wa[k0 + khalf + 0];
    a.y  = Wa[k0 + khalf + 1];
    bf.x = Sb[(size_t)(k0 + khalf + 0) * NHW];
    bf.y = Sb[(size_t)(k0 + khalf + 1) * NHW];
    // D = A(16x4 f32) x B(4x16 f32) + C   -> v_wmma_f32_16x16x4_f32
    acc = __builtin_amdgcn_wmma_f32_16x16x4_f32(
        /*neg_a=*/false, a, /*neg_b=*/false, bf,
        /*c_mod=*/(short)0, acc, /*reuse_a=*/false, /*reuse_b=*/false);
  }

  const int mhi = (lane >> 4) << 3;          // 0 or 8
  // out strides: b:25165824  l:8388608  p:262144  sp:32768  j:4096
#pragma unroll
  for (int r = 0; r < 8; ++r) {
    int o = m0 + r + mhi;                    // output channel 0..63
    float v = acc[r] + bias[o];
    int sp = o >> 3, j = o & 7;
    size_t base = (size_t)b * 25165824u + (size_t)l * 8388608u
                + (size_t)sp * 32768u + (size_t)j * 4096u + (size_t)pix;
#pragma unroll 4
    for (int p = 0; p < NP; ++p) {
      __builtin_nontemporal_store(v, out + base + (size_t)p * 262144u);
    }
  }
}

extern "C" void kernel_launch(void* const* d_in, const int* in_sizes, int n_in,
                              void* d_out, int out_size, void* d_ws, size_t ws_size,
                              hipStream_t stream) {
  const float* feat0  = (const float*)d_in[0];
  const float* feat1  = (const float*)d_in[1];
  const float* feat2  = (const float*)d_in[2];
  // d_in[3] = grid : unused by the math (only provides P via broadcast)
  const float* w_off  = (const float*)d_in[4];
  const float* b_off  = (const float*)d_in[5];
  const float* weight = (const float*)d_in[6];
  const float* bias   = (const float*)d_in[7];
  float* out = (float*)d_out;

  float* off_ws = (float*)d_ws;              // B*16*HW       = 131072 f32 (0.5 MB)
  float* S      = off_ws + NB * 16 * NHW;    // L*B*C*HW      = 1572864 f32 (6.3 MB)

  edp_offset_conv  <<<(NB * NHW) / 256, 256, 0, stream>>>(feat0, w_off, b_off, off_ws);
  edp_deform_sample<<<(NL * NB * NSP * NHW) / 256, 256, 0, stream>>>(feat0, feat1, feat2, off_ws, S);
  edp_gemm_wmma    <<<NL * NB * 256, 128, 0, stream>>>(S, weight, bias, out);
}